// Fourier_9844065042502
// MI455X (gfx1250) — compile-verified
//
#include <hip/hip_runtime.h>
#include <stdint.h>

#define CCH 2048
#define THREADS 256
#define PI2 6.28318530717958647692f

typedef __attribute__((ext_vector_type(2))) float v2f;
typedef __attribute__((ext_vector_type(8))) float v8f;

// Fused Re(FFT_2048(x)) + LayerNorm, one workgroup (8 wave32) per row.
// 2048 = 16 (n1, radix-16 combine via WMMA) x 128 (n2, Stockham FFT on VALU).
__global__ __launch_bounds__(THREADS) void fourier_ln_kernel(
    const float* __restrict__ x, const float* __restrict__ gamma,
    const float* __restrict__ beta, float* __restrict__ out, int nrows)
{
    __shared__ float Are[CCH], Aim[CCH], Bre[CCH], Bim[CCH];
    __shared__ float twr[64], twi[64];
    __shared__ float redA[8], redB[8];
    __shared__ float s_mu, s_rs;

    const int tid = threadIdx.x;
    const int row = blockIdx.x;
    if (row >= nrows) return;
    const float* xrow = x + (size_t)row * CCH;

    // ---- async copy row (8KB) into Are via CDNA5 async-to-LDS path ----
    {
        unsigned lds0 = (unsigned)(uintptr_t)(&Are[0]);
        unsigned long long sbase = (unsigned long long)(uintptr_t)xrow;
        for (int i = tid; i < CCH / 4; i += THREADS) {
            unsigned voff = (unsigned)(i * 16);
            unsigned ldsa = lds0 + voff;
            asm volatile("global_load_async_to_lds_b128 %0, %1, %2"
                         :: "v"(ldsa), "v"(voff), "s"(sbase) : "memory");
        }
    }
    // gamma/beta for this thread's contiguous 8-element output chunk:
    // issue early so the loads overlap the FFT stages.
    const int c0 = tid * 8;
    const float4 g0 = *(const float4*)(gamma + c0);
    const float4 g1 = *(const float4*)(gamma + c0 + 4);
    const float4 b0 = *(const float4*)(beta + c0);
    const float4 b1 = *(const float4*)(beta + c0 + 4);

    // imag part starts at zero
    for (int i = tid; i < CCH; i += THREADS) Aim[i] = 0.0f;
    // twiddle table for 128-pt FFT: tw[p] = exp(-2*pi*i*p/128), p<64
    if (tid < 64) {
        float s, c;
        __sincosf(-PI2 * (float)tid / 128.0f, &s, &c);
        twr[tid] = c; twi[tid] = s;
    }
    asm volatile("s_wait_asynccnt 0" ::: "memory");
    __syncthreads();

    // ---- stage 1: 16 batched 128-pt Stockham FFTs over n2 ----
    // logical element (n1, p) lives at LDS index n1 + 16*p  (== natural row order)
    float *are = Are, *aim = Aim, *bre = Bre, *bim = Bim;
    int m = 1;
    #pragma unroll 1
    for (int stage = 0; stage < 7; ++stage) {
        for (int t = tid; t < 1024; t += THREADS) {
            int n1 = t & 15;
            int tb = t >> 4;            // butterfly index in [0,64)
            int k  = tb & (m - 1);
            int jm = tb - k;            // j*m
            int i0 = n1 + 16 * tb;
            int i1 = n1 + 16 * (tb + 64);
            float c0r = are[i0], c0i = aim[i0];
            float c1r = are[i1], c1i = aim[i1];
            float wr = twr[jm], wi = twi[jm];
            float dr = c0r - c1r, di = c0i - c1i;
            int o0 = n1 + 16 * (tb + jm);
            int o1 = o0 + 16 * m;
            bre[o0] = c0r + c1r;  bim[o0] = c0i + c1i;
            bre[o1] = wr * dr - wi * di;
            bim[o1] = wr * di + wi * dr;
        }
        __syncthreads();
        float* tr;
        tr = are; are = bre; bre = tr;
        tr = aim; aim = bim; bim = tr;
        m <<= 1;
    }
    // now (are,aim) hold G[n1][k2] at index n1 + 16*k2, k2 natural order

    // ---- twiddle: G'[n1][k2] = G[n1][k2] * exp(-2*pi*i*n1*k2/2048) ----
    for (int i = tid; i < CCH; i += THREADS) {
        int n1 = i & 15, k2 = i >> 4;
        float s, c;
        __sincosf(-PI2 * (float)(n1 * k2) / 2048.0f, &s, &c);
        float gr = are[i], gi = aim[i];
        are[i] = c * gr - s * gi;
        aim[i] = c * gi + s * gr;
    }
    __syncthreads();

    // ---- stage 2 (radix-16 real-output combine) on the matrix pipe ----
    // Y[k1][k2] = sum_n1 cos(2pi n1 k1/16)*G'r + sin(2pi n1 k1/16)*G'i
    // One 16(k1) x 16(k2) output tile per wave; 8 waves cover k2 in [0,128).
    {
        const int lane = tid & 31;
        const int w    = tid >> 5;          // wave id 0..7 -> k2 block
        const int nlo  = lane & 15;         // M (=k1) for A, N (=k2 in tile) for B/C
        const int hi   = lane >> 4;         // K half selector per documented A layout
        const int k2   = w * 16 + nlo;
        v8f acc = {};
        #pragma unroll
        for (int kk = 0; kk < 16; kk += 4) {
            int ka = kk + hi * 2;           // K index for .x
            int kb = ka + 1;                // K index for .y
            v2f ac, as2, br, bi;
            {
                float a0 = (PI2 / 16.0f) * (float)(nlo * ka);
                float a1 = (PI2 / 16.0f) * (float)(nlo * kb);
                ac.x = __cosf(a0);  ac.y = __cosf(a1);
                as2.x = __sinf(a0); as2.y = __sinf(a1);
            }
            br.x = are[ka + 16 * k2];  br.y = are[kb + 16 * k2];
            bi.x = aim[ka + 16 * k2];  bi.y = aim[kb + 16 * k2];
            acc = __builtin_amdgcn_wmma_f32_16x16x4_f32(
                false, ac, false, br, (short)0, acc, false, false);
            acc = __builtin_amdgcn_wmma_f32_16x16x4_f32(
                false, as2, false, bi, (short)0, acc, false, false);
        }
        // C/D layout: VGPR r -> M = r + hi*8, N = lane&15
        #pragma unroll
        for (int r = 0; r < 8; ++r) {
            int k1 = r + hi * 8;
            bre[k2 + 128 * k1] = acc[r];    // y (real FFT output) in bre
        }
    }
    __syncthreads();

    // ---- LayerNorm over the row (y kept in registers across the barrier) ----
    float4 y0 = *(const float4*)(bre + c0);        // ds_load_b128
    float4 y1 = *(const float4*)(bre + c0 + 4);
    float s1 = (y0.x + y0.y) + (y0.z + y0.w) + (y1.x + y1.y) + (y1.z + y1.w);
    float s2 = (y0.x * y0.x + y0.y * y0.y) + (y0.z * y0.z + y0.w * y0.w) +
               (y1.x * y1.x + y1.y * y1.y) + (y1.z * y1.z + y1.w * y1.w);
    #pragma unroll
    for (int off = 16; off > 0; off >>= 1) {
        s1 += __shfl_down(s1, off, 32);
        s2 += __shfl_down(s2, off, 32);
    }
    if ((tid & 31) == 0) { redA[tid >> 5] = s1; redB[tid >> 5] = s2; }
    __syncthreads();
    if (tid == 0) {
        float a = 0.0f, b = 0.0f;
        #pragma unroll
        for (int i = 0; i < 8; ++i) { a += redA[i]; b += redB[i]; }
        float mu  = a * (1.0f / (float)CCH);
        float var = b * (1.0f / (float)CCH) - mu * mu;
        s_mu = mu;
        s_rs = rsqrtf(var + 1e-5f);
    }
    __syncthreads();
    const float mu = s_mu, rs = s_rs;
    float* orow = out + (size_t)row * CCH;
    float4 o0, o1;
    o0.x = (y0.x - mu) * rs * g0.x + b0.x;
    o0.y = (y0.y - mu) * rs * g0.y + b0.y;
    o0.z = (y0.z - mu) * rs * g0.z + b0.z;
    o0.w = (y0.w - mu) * rs * g0.w + b0.w;
    o1.x = (y1.x - mu) * rs * g1.x + b1.x;
    o1.y = (y1.y - mu) * rs * g1.y + b1.y;
    o1.z = (y1.z - mu) * rs * g1.z + b1.z;
    o1.w = (y1.w - mu) * rs * g1.w + b1.w;
    *(float4*)(orow + c0)     = o0;                // global_store_b128
    *(float4*)(orow + c0 + 4) = o1;
}

extern "C" void kernel_launch(void* const* d_in, const int* in_sizes, int n_in,
                              void* d_out, int out_size, void* d_ws, size_t ws_size,
                              hipStream_t stream) {
    const float* x     = (const float*)d_in[0];
    const float* gamma = (const float*)d_in[1];
    const float* beta  = (const float*)d_in[2];
    float* out = (float*)d_out;
    int nrows = out_size / CCH;   // 4 * 4096 = 16384
    dim3 grid((unsigned)nrows), block(THREADS);
    hipLaunchKernelGGL(fourier_ln_kernel, grid, block, 0, stream,
                       x, gamma, beta, out, nrows);
}